// Dynamic_conv_7799660609550
// MI455X (gfx1250) — compile-verified
//
#include <hip/hip_runtime.h>
#include <cstdint>
#include <cstddef>

// Dynamic 3x3 convolution with per-pixel softmax weights (reflect padding).
// x:      (4, 64, 256, 256) f32
// kernel: (4,  9, 256, 256) f32
// out:    (4, 64, 256, 256) f32
//
// Memory-bound (AI ~= 2 FLOP/byte): optimized around the CDNA5 async
// global->LDS path (ASYNCcnt) with a double-buffered channel pipeline.

#define BATCH 4
#define CHAN  64
#define HH    256
#define WW    256
#define HWSZ  (HH * WW)

typedef __attribute__((address_space(1))) int* gbl_i32_ptr;
typedef __attribute__((address_space(3))) int* lds_i32_ptr;

// ---------------- CDNA5 async global->LDS copy (ASYNCcnt-tracked) ----------
__device__ __forceinline__ void async_cp_b32(void* lds_dst, const void* gsrc) {
#if __has_builtin(__builtin_amdgcn_global_load_async_to_lds_b32)
  __builtin_amdgcn_global_load_async_to_lds_b32(
      (gbl_i32_ptr)gsrc,        // global source (param type: __device__ int*)
      (lds_i32_ptr)lds_dst,     // LDS destination
      /*imm offset*/ 0, /*cpol*/ 0);
#else
  // GV mode: LDS address comes from the vdst VGPR; global address from a 64-bit VGPR pair.
  asm volatile("global_load_async_to_lds_b32 %0, %1, off"
               :
               : "v"((uint32_t)(uintptr_t)lds_dst), "v"(gsrc)
               : "memory");
#endif
}

__device__ __forceinline__ void wait_async0() {
#if __has_builtin(__builtin_amdgcn_s_wait_asynccnt)
  __builtin_amdgcn_s_wait_asynccnt(0);
#else
  asm volatile("s_wait_asynccnt 0" ::: "memory");
#endif
}

__device__ __forceinline__ int reflect_idx(int i, int n) {
  i = (i < 0) ? -i : i;
  i = (i >= n) ? (2 * n - 2 - i) : i;
  return i;
}

__global__ __launch_bounds__(WW)
void dynconv3x3_kernel(const float* __restrict__ x,
                       const float* __restrict__ kern,
                       float* __restrict__ out) {
  // Double-buffered 3-row strip of one channel: 2 * 3 * 256 * 4B = 6 KB LDS.
  __shared__ float tile[2][3][WW];

  const int w = threadIdx.x;            // 0..255 (one pixel column per lane)
  const int h = blockIdx.x & (HH - 1);  // row
  const int b = blockIdx.x >> 8;        // batch (HH == 256)

  // ---- per-pixel softmax over the 9 logits (shared across all 64 channels)
  const float* kp = kern + (size_t)b * 9 * HWSZ + (size_t)h * WW + w;
  float wgt[9];
  float m = -INFINITY;
#pragma unroll
  for (int t = 0; t < 9; ++t) {
    wgt[t] = kp[(size_t)t * HWSZ];
    m = fmaxf(m, wgt[t]);
  }
  float s = 0.0f;
#pragma unroll
  for (int t = 0; t < 9; ++t) {
    wgt[t] = __expf(wgt[t] - m);
    s += wgt[t];
  }
  const float inv = 1.0f / s;
#pragma unroll
  for (int t = 0; t < 9; ++t) wgt[t] *= inv;

  // Reflected source rows/cols ('reflect' mode: -1 -> 1, N -> N-2).
  const int hr0 = reflect_idx(h - 1, HH);
  const int hr1 = h;
  const int hr2 = reflect_idx(h + 1, HH);
  const int wl  = reflect_idx(w - 1, WW);
  const int wr  = reflect_idx(w + 1, WW);

  const float* xb = x + (size_t)b * CHAN * HWSZ;
  float*       op = out + (size_t)b * CHAN * HWSZ + (size_t)h * WW + w;

  // ---- prologue: stage channel 0 into buffer 0 via async copies
  {
    const float* xc = xb;  // c = 0
    async_cp_b32(&tile[0][0][w], xc + (size_t)hr0 * WW + w);
    async_cp_b32(&tile[0][1][w], xc + (size_t)hr1 * WW + w);
    async_cp_b32(&tile[0][2][w], xc + (size_t)hr2 * WW + w);
  }
  wait_async0();
  __syncthreads();

  // ---- channel pipeline: prefetch c+1 while computing c
#pragma unroll 1
  for (int c = 0; c < CHAN; ++c) {
    const int cur = c & 1;
    const int nxt = cur ^ 1;

    if (c + 1 < CHAN) {
      const float* xc = xb + (size_t)(c + 1) * HWSZ;
      async_cp_b32(&tile[nxt][0][w], xc + (size_t)hr0 * WW + w);
      async_cp_b32(&tile[nxt][1][w], xc + (size_t)hr1 * WW + w);
      async_cp_b32(&tile[nxt][2][w], xc + (size_t)hr2 * WW + w);
    }

    // 9-tap weighted sum out of LDS (ds_load_b32, neighbor reuse in LDS).
    float acc;
    acc  = wgt[0] * tile[cur][0][wl];
    acc  = fmaf(wgt[1], tile[cur][0][w],  acc);
    acc  = fmaf(wgt[2], tile[cur][0][wr], acc);
    acc  = fmaf(wgt[3], tile[cur][1][wl], acc);
    acc  = fmaf(wgt[4], tile[cur][1][w],  acc);
    acc  = fmaf(wgt[5], tile[cur][1][wr], acc);
    acc  = fmaf(wgt[6], tile[cur][2][wl], acc);
    acc  = fmaf(wgt[7], tile[cur][2][w],  acc);
    acc  = fmaf(wgt[8], tile[cur][2][wr], acc);

    op[(size_t)c * HWSZ] = acc;  // coalesced B32 store

    wait_async0();   // this wave's prefetch of tile[nxt] landed in LDS
    __syncthreads(); // all waves done reading tile[cur] / writing tile[nxt]
  }
}

extern "C" void kernel_launch(void* const* d_in, const int* in_sizes, int n_in,
                              void* d_out, int out_size, void* d_ws, size_t ws_size,
                              hipStream_t stream) {
  (void)in_sizes; (void)n_in; (void)out_size; (void)d_ws; (void)ws_size;
  const float* x    = (const float*)d_in[0];
  const float* kern = (const float*)d_in[1];
  float*       out  = (float*)d_out;

  dim3 grid(BATCH * HH);  // 1024 blocks: one (batch, row) strip each
  dim3 block(WW);         // 256 threads = 8 wave32
  dynconv3x3_kernel<<<grid, block, 0, stream>>>(x, kern, out);
}